// BOJANET_23545010717406
// MI455X (gfx1250) — compile-verified
//
#include <hip/hip_runtime.h>

// ---------------------------------------------------------------------------
// BOJANET fused demod + gated recurrence for MI455X (gfx1250, wave32, WMMA)
//
//   grid  = B/16 = 32 workgroups (one 16-row batch tile each, ~one per WGP)
//   block = 128 threads = 4 waves: wave0 = serial-scan consumer,
//                                  waves1-3 = FIR/demod/projection producers
//   Producers fill a double-buffered LDS ring (Lf, Lg, sin, cos per step);
//   consumer runs the recurrence with chained fp32 WMMA 16x16x4 matvecs.
//
// GEMM orientation (channel-major):  D[ch, batch] = W[ch, k] * X[k, batch]
//   A (16x4 f32): lane m=l&15 holds W[m, k0+2h], W[m, k0+2h+1]
//   B (4x16 f32): lane holds X[k0+2h, n], X[k0+2h+1, n], n=l&15
//   C/D (16x16) : vgpr r, lane l -> [ch = r+8h, batch n]
// All LDS tiles are [batch][16ch] so a lane's 8 D values are contiguous
// (b128 DS traffic) and all channel dims are padded to 16 (zero rows) so the
// hot loops have no divergent guards (EXEC stays all-ones around WMMA).
// ---------------------------------------------------------------------------

#define WINDOW  16
#define VD      6
#define HIDDEN  12
#define SEQ     8192
#define BATCH   512
#define TCHUNK  10          // ring steps per phase (static LDS = 62 KB)
#define EPSF    1e-8f

typedef __attribute__((ext_vector_type(2))) float v2f;
typedef __attribute__((ext_vector_type(8))) float v8f;

static __device__ __forceinline__ v8f wmma4(v2f a, v2f b, v8f c) {
    // V_WMMA_F32_16X16X4_F32 : D = A(16x4) x B(4x16) + C, full fp32
    return __builtin_amdgcn_wmma_f32_16x16x4_f32(
        false, a, false, b, (short)0, c, false, false);
}

__global__ __launch_bounds__(128, 1)
void bojanet_fused(const float2* __restrict__ x2,       // (B,S) of (I,Q)
                   const float*  __restrict__ firI,     // (6,16)
                   const float*  __restrict__ firQ,     // (6,16)
                   const float*  __restrict__ Wfi,      // (12,12)
                   const float*  __restrict__ bfi,      // (12)
                   const float*  __restrict__ Wfh,      // (12,12)
                   const float*  __restrict__ Wgi,      // (12,12)
                   const float*  __restrict__ bgi,      // (12)
                   const float*  __restrict__ Wgh,      // (12,12)
                   const float*  __restrict__ WoI,      // (1,12)
                   const float*  __restrict__ boI,      // (1)
                   const float*  __restrict__ WoQ,      // (1,12)
                   const float*  __restrict__ boQ,      // (1)
                   float2*       __restrict__ out2)     // (B,S) of (a-b, b+a)
{
    // ---- LDS (all [batch][channel], channel padded to 16) -----------------
    __shared__ float ringLf [2][TCHUNK][16][16];
    __shared__ float ringLg [2][TCHUNK][16][16];
    __shared__ float ringSin[2][TCHUNK][16][ 6];
    __shared__ float ringCos[2][TCHUNK][16][ 6];
    __shared__ float sH[16][16];          // h as [batch][k] for B operands
    __shared__ float prodFir[3][16][16];  // per-producer-wave scratch
    __shared__ float prodL  [3][16][16];

    const int wave = threadIdx.x >> 5;
    const int lane = threadIdx.x & 31;
    const int hh   = lane >> 4;          // lane half (0/1)
    const int n    = lane & 15;          // batch column within tile
    const int m    = lane & 15;          // A-matrix row (channel)
    const int row0 = blockIdx.x * 16;
    const int nchunks = (SEQ + TCHUNK - 1) / TCHUNK;

    // ---- Preload A operands & per-lane constants --------------------------
    v2f a1[4], a2[4];                     // FIR combos: rows 0-5 I_fir, 6-11 Q_fir
    for (int j = 0; j < 4; ++j) {
        const int k = 4*j + 2*hh;
        float x1 = 0.f, y1 = 0.f, x2v = 0.f, y2 = 0.f;
        if (m < 12) {
            if (m < 6) { x1 =  firI[m*16+k];      y1 =  firI[m*16+k+1];
                         x2v = -firQ[m*16+k];     y2 = -firQ[m*16+k+1]; }
            else       { x1 =  firQ[(m-6)*16+k];  y1 =  firQ[(m-6)*16+k+1];
                         x2v =  firI[(m-6)*16+k]; y2 =  firI[(m-6)*16+k+1]; }
        }
        a1[j].x = x1;  a1[j].y = y1;
        a2[j].x = x2v; a2[j].y = y2;
    }
    v2f afi[3], agi[3], afh[3], agh[3];
    for (int j = 0; j < 3; ++j) {
        const int k = 4*j + 2*hh;
        float fx=0,fy=0,gx=0,gy=0,fhx=0,fhy=0,ghx=0,ghy=0;
        if (m < 12) {
            fx  = Wfi[m*12+k]; fy  = Wfi[m*12+k+1];
            gx  = Wgi[m*12+k]; gy  = Wgi[m*12+k+1];
            fhx = Wfh[m*12+k]; fhy = Wfh[m*12+k+1];
            ghx = Wgh[m*12+k]; ghy = Wgh[m*12+k+1];
        }
        afi[j].x=fx;  afi[j].y=fy;  agi[j].x=gx;  agi[j].y=gy;
        afh[j].x=fhx; afh[j].y=fhy; agh[j].x=ghx; agh[j].y=ghy;
    }
    v8f cbf, cbg;
    float wOutI[8], wOutQ[8];
    #pragma unroll
    for (int r = 0; r < 8; ++r) {
        const int cch = r + 8*hh;
        cbf[r]   = (cch < 12) ? bfi[cch] : 0.f;
        cbg[r]   = (cch < 12) ? bgi[cch] : 0.f;
        wOutI[r] = (cch < 12) ? WoI[cch] : 0.f;
        wOutQ[r] = (cch < 12) ? WoQ[cch] : 0.f;
    }
    const float biasI = boI[0], biasQ = boQ[0];

    // ---- Producer: fill ring for one chunk --------------------------------
    auto fill_chunk = [&](int chunk, int buf) {
        const int w = wave - 1;                       // 0..2
        float* pFir = &prodFir[w][0][0];
        float* pL   = &prodL  [w][0][0];
        for (int i = w; i < TCHUNK; i += 3) {         // wave-uniform t
            const int t = chunk * TCHUNK + i;
            if (t >= SEQ) continue;
            float* rLf  = &ringLf [buf][i][0][0];
            float* rLg  = &ringLg [buf][i][0][0];
            float* rSin = &ringSin[buf][i][0][0];
            float* rCos = &ringCos[buf][i][0][0];

            // window samples: B operand [tap][batch]; (I,Q) adjacent -> b64
            const float2* px = x2 + (size_t)(row0+n)*SEQ + (t - 15 + 2*hh);
            float2 s[8];
            if (t >= 15) {                            // steady state: no guards
                #pragma unroll
                for (int j = 0; j < 4; ++j) { s[2*j] = px[4*j]; s[2*j+1] = px[4*j+1]; }
            } else {                                  // sequence head only
                #pragma unroll
                for (int j = 0; j < 4; ++j) {
                    const int t0 = t - 15 + 2*hh + 4*j;
                    s[2*j]   = (t0   >= 0) ? px[4*j]   : make_float2(0.f, 0.f);
                    s[2*j+1] = (t0+1 >= 0) ? px[4*j+1] : make_float2(0.f, 0.f);
                }
            }
            // FIR: D[ch,batch] = A1 @ Iwin + A2 @ Qwin   (K=16 as 4 chunks)
            v8f d = {};
            #pragma unroll
            for (int j = 0; j < 4; ++j) {
                v2f bi; bi.x = s[2*j].x; bi.y = s[2*j+1].x;
                v2f bq; bq.x = s[2*j].y; bq.y = s[2*j+1].y;
                d = wmma4(a1[j], bi, d);
                d = wmma4(a2[j], bq, d);
            }
            *(float4*)(pFir + n*16 + 8*hh)     = make_float4(d[0],d[1],d[2],d[3]);
            *(float4*)(pFir + n*16 + 8*hh + 4) = make_float4(d[4],d[5],d[6],d[7]);
            __builtin_amdgcn_wave_barrier();
            // Demod: 96 (u,n) pairs over 32 lanes, 3 each
            #pragma unroll
            for (int q = 0; q < 3; ++q) {
                const int id = lane + 32*q;           // 0..95
                const int u  = id >> 4;               // 0..5
                const int n2 = id & 15;
                const float If  = pFir[n2*16 + u];
                const float Qf  = pFir[n2*16 + u + 6];
                const float mag = __builtin_amdgcn_sqrtf(If*If + Qf*Qf) + EPSF;
                const float rm  = __builtin_amdgcn_rcpf(mag);
                pL[n2*16 + u]     = mag;
                pL[n2*16 + u + 6] = mag * mag;
                rCos[n2*6 + u] = If * rm;
                rSin[n2*6 + u] = Qf * rm;
            }
            __builtin_amdgcn_wave_barrier();
            // Input projections: Lf/Lg[ch,batch] = Wfi/Wgi @ L + bias
            v8f df = cbf, dg = cbg;
            #pragma unroll
            for (int j = 0; j < 3; ++j) {
                const int k = 4*j + 2*hh;             // <= 11
                const float2 lv = *(const float2*)(pL + n*16 + k);
                v2f bl; bl.x = lv.x; bl.y = lv.y;
                df = wmma4(afi[j], bl, df);
                dg = wmma4(agi[j], bl, dg);
            }
            *(float4*)(rLf + n*16 + 8*hh)     = make_float4(df[0],df[1],df[2],df[3]);
            *(float4*)(rLf + n*16 + 8*hh + 4) = make_float4(df[4],df[5],df[6],df[7]);
            *(float4*)(rLg + n*16 + 8*hh)     = make_float4(dg[0],dg[1],dg[2],dg[3]);
            *(float4*)(rLg + n*16 + 8*hh + 4) = make_float4(dg[4],dg[5],dg[6],dg[7]);
        }
    };

    // ---- Consumer: serial recurrence over one chunk -----------------------
    v8f hD = {};                                       // h in D layout
    auto consume_chunk = [&](int chunk, int buf) {
        for (int i = 0; i < TCHUNK; ++i) {
            const int t = chunk * TCHUNK + i;
            if (t >= SEQ) return;
            const float* rLf  = &ringLf [buf][i][0][0];
            const float* rLg  = &ringLg [buf][i][0][0];
            const float* rSin = &ringSin[buf][i][0][0];
            const float* rCos = &ringCos[buf][i][0][0];
            // C init = Lf / Lg from ring (2x b128 each, rows 12-15 are zero)
            const float4 f0 = *(const float4*)(rLf + n*16 + 8*hh);
            const float4 f1 = *(const float4*)(rLf + n*16 + 8*hh + 4);
            const float4 g0 = *(const float4*)(rLg + n*16 + 8*hh);
            const float4 g1 = *(const float4*)(rLg + n*16 + 8*hh + 4);
            v8f pf, pg;
            pf[0]=f0.x; pf[1]=f0.y; pf[2]=f0.z; pf[3]=f0.w;
            pf[4]=f1.x; pf[5]=f1.y; pf[6]=f1.z; pf[7]=f1.w;
            pg[0]=g0.x; pg[1]=g0.y; pg[2]=g0.z; pg[3]=g0.w;
            pg[4]=g1.x; pg[5]=g1.y; pg[6]=g1.z; pg[7]=g1.w;
            // + Wfh @ h, + Wgh @ h  (K=12 as 3 chained K=4 WMMAs, shared B)
            #pragma unroll
            for (int j = 0; j < 3; ++j) {
                const int k = 4*j + 2*hh;             // <= 11
                const float2 hv = *(const float2*)(&sH[0][0] + n*16 + k);
                v2f bh; bh.x = hv.x; bh.y = hv.y;
                pf = wmma4(afh[j], bh, pf);
                pg = wmma4(agh[j], bh, pg);
            }
            __builtin_amdgcn_wave_barrier();          // B reads before h store
            // activations + state update + phase-rotated output partials
            float pa = 0.f, pb = 0.f;
            float hs[8];
            #pragma unroll
            for (int r = 0; r < 8; ++r) {
                const int cch = r + 8*hh;
                const int u   = cch % 6;              // cos_c/sin_c index
                const float f  = __builtin_amdgcn_rcpf(1.f + __expf(-pf[r]));
                const float e2 = __expf(2.f * pg[r]);
                const float g  = 1.f - 2.f * __builtin_amdgcn_rcpf(e2 + 1.f);
                const float hn = f * hD[r] + (1.f - f) * g;
                hD[r] = hn; hs[r] = hn;
                pa += hn * rCos[n*6 + u] * wOutI[r];  // wOut rows 12-15 are 0
                pb += hn * rSin[n*6 + u] * wOutQ[r];
            }
            *(float4*)(&sH[0][0] + n*16 + 8*hh)     = make_float4(hs[0],hs[1],hs[2],hs[3]);
            *(float4*)(&sH[0][0] + n*16 + 8*hh + 4) = make_float4(hs[4],hs[5],hs[6],hs[7]);
            __builtin_amdgcn_wave_barrier();
            // reduce channel halves across lane halves
            pa += __shfl_xor(pa, 16, 32);
            pb += __shfl_xor(pb, 16, 32);
            if (lane < 16) {
                const float a = pa + biasI, b = pb + biasQ;
                out2[(size_t)(row0+n)*SEQ + t] = make_float2(a - b, b + a);
            }
        }
    };

    // ---- Pipeline ---------------------------------------------------------
    if (wave == 0) {
        for (int idx = lane; idx < 256; idx += 32) (&sH[0][0])[idx] = 0.f;
    } else {
        fill_chunk(0, 0);                              // prologue: chunk 0
    }
    __syncthreads();
    for (int c = 0; c < nchunks; ++c) {
        if (wave == 0) {
            consume_chunk(c, c & 1);
        } else if (c + 1 < nchunks) {
            fill_chunk(c + 1, (c + 1) & 1);
        }
        __syncthreads();
    }
}

extern "C" void kernel_launch(void* const* d_in, const int* in_sizes, int n_in,
                              void* d_out, int out_size, void* d_ws, size_t ws_size,
                              hipStream_t stream) {
    (void)in_sizes; (void)n_in; (void)out_size; (void)d_ws; (void)ws_size;
    const float2* x2   = (const float2*)d_in[0];
    const float*  firI = (const float*)d_in[1];
    const float*  firQ = (const float*)d_in[2];
    const float*  Wfi  = (const float*)d_in[3];
    const float*  bfi  = (const float*)d_in[4];
    const float*  Wfh  = (const float*)d_in[5];
    const float*  Wgi  = (const float*)d_in[6];
    const float*  bgi  = (const float*)d_in[7];
    const float*  Wgh  = (const float*)d_in[8];
    const float*  WoI  = (const float*)d_in[9];
    const float*  boI  = (const float*)d_in[10];
    const float*  WoQ  = (const float*)d_in[11];
    const float*  boQ  = (const float*)d_in[12];
    float2* out2 = (float2*)d_out;

    dim3 grid(BATCH / 16), block(128);
    hipLaunchKernelGGL(bojanet_fused, grid, block, 0, stream,
                       x2, firI, firQ, Wfi, bfi, Wfh, Wgi, bgi, Wgh,
                       WoI, boI, WoQ, boQ, out2);
}